// TemporalSelection_19009525252355
// MI455X (gfx1250) — compile-verified
//
#include <hip/hip_runtime.h>
#include <math.h>

#define BB 32
#define TT 4096
#define CC 512
#define HH 4
#define DHD 128

typedef float v2f __attribute__((ext_vector_type(2)));
typedef float v8f __attribute__((ext_vector_type(8)));
typedef int v4i_vs __attribute__((vector_size(16)));  // matches builtin param type

#if defined(__has_builtin)
#if __has_builtin(__builtin_amdgcn_global_load_async_to_lds_b128)
#define HAVE_ASYNC_LDS 1
#endif
#endif

#ifdef HAVE_ASYNC_LDS
#define GAS1 __attribute__((address_space(1)))
#define LAS3 __attribute__((address_space(3)))
#endif

// ---------------------------------------------------------------------------
// Kernel 1: q = key @ w_q^T + b_q  (B x C), then fold w_k into the q side:
//   vq[b,h,c] = sum_d q[b,h*DH+d] * w_k[h*DH+d, c]
//   c0[b,h]   = sum_d q[b,h*DH+d] * b_k[h*DH+d]
// ---------------------------------------------------------------------------
__global__ __launch_bounds__(256) void ts_proj_kernel(
    const float* __restrict__ key, const float* __restrict__ w_q,
    const float* __restrict__ b_q, const float* __restrict__ w_k,
    const float* __restrict__ b_k, float* __restrict__ vq,
    float* __restrict__ c0) {
  __shared__ float key_s[CC];
  __shared__ float q_s[CC];
  int b = blockIdx.x, tid = threadIdx.x;
  for (int c = tid; c < CC; c += 256) key_s[c] = key[(size_t)b * CC + c];
  __syncthreads();
  for (int j = tid; j < CC; j += 256) {
    float s = b_q[j];
    const float* wr = w_q + (size_t)j * CC;
    for (int c = 0; c < CC; ++c) s += key_s[c] * wr[c];
    q_s[j] = s;
  }
  __syncthreads();
  for (int o = tid; o < HH * CC; o += 256) {
    int h = o >> 9, c = o & (CC - 1);
    float s = 0.f;
    for (int d = 0; d < DHD; ++d)
      s += q_s[h * DHD + d] * w_k[(size_t)(h * DHD + d) * CC + c];
    vq[(size_t)b * HH * CC + o] = s;
  }
  if (tid < HH) {
    float s = 0.f;
    for (int d = 0; d < DHD; ++d) s += q_s[tid * DHD + d] * b_k[tid * DHD + d];
    c0[b * HH + tid] = s;
  }
}

// ---------------------------------------------------------------------------
// Kernel 2: scores[b,h,t] = (vq[b,h,:]·query[b,t,:] + c0[b,h]) / sqrt(DH)
// via V_WMMA_F32_16X16X4_F32. Query staged through LDS with
// GLOBAL_LOAD_ASYNC_TO_LDS_B128 (ASYNCcnt) for full MLP; B fragments read
// branch-free via a zero-row pointer for lanes n>=4.
// ---------------------------------------------------------------------------
#define ROWS 128
#define KCH 64
#define LDSROW (KCH + 4)  // pad: bank = (4*m + c) % 64, conflict-free

__global__ __launch_bounds__(256) void ts_scores_kernel(
    const float* __restrict__ query, const float* __restrict__ vq,
    const float* __restrict__ c0, float* __restrict__ scores) {
  __shared__ float vq_s[HH * CC];
  __shared__ float zbuf[CC];       // all-zero B rows for lanes n >= HH
  __shared__ float c0_s[HH];
  __shared__ float qt[ROWS * LDSROW];
  int tid = threadIdx.x;
  int b = blockIdx.x >> 5;                 // 32 t-tiles per batch
  int tbase = (blockIdx.x & 31) * ROWS;

  for (int o = tid; o < HH * CC; o += 256) vq_s[o] = vq[(size_t)b * HH * CC + o];
  for (int o = tid; o < CC; o += 256) zbuf[o] = 0.f;
  if (tid < HH) c0_s[tid] = c0[b * HH + tid];

  int wave = tid >> 5, lane = tid & 31;
  int m = lane & 15;                // A row (M)
  int kb = (lane >> 4) << 1;        // K pair base: lanes 0-15 -> K0,1 ; 16-31 -> K2,3
  int n = lane & 15;                // B/D column (N); heads live in n<4
  const float* bptr = (n < HH) ? (vq_s + n * CC) : zbuf;  // branch-free B source
  v8f acc = {};

  const float* qbase = query + ((size_t)b * TT + tbase) * CC;

  for (int ch = 0; ch < CC; ch += KCH) {
    __syncthreads();  // LDS tile reuse fence (also covers initial vq_s/zbuf fill)
#ifdef HAVE_ASYNC_LDS
    // 8 x 16B async copies per thread: issue all, one ASYNCcnt wait at the end.
    #pragma unroll
    for (int i = 0; i < 8; ++i) {
      int idx = tid + i * 256;
      int r = idx >> 4, q4 = (idx & 15) << 2;
      const float* src = qbase + (size_t)r * CC + ch + q4;
      float* dst = &qt[r * LDSROW + q4];
      __builtin_amdgcn_global_load_async_to_lds_b128(
          (GAS1 v4i_vs*)src, (LAS3 v4i_vs*)dst, 0, 0);
    }
#if __has_builtin(__builtin_amdgcn_s_wait_asynccnt)
    __builtin_amdgcn_s_wait_asynccnt(0);
#else
    asm volatile("s_wait_asynccnt 0x0" ::: "memory");
#endif
#else
    // Fallback: stage through VGPRs, all loads issued before any LDS store.
    float4 tmp[8];
    #pragma unroll
    for (int i = 0; i < 8; ++i) {
      int idx = tid + i * 256;
      int r = idx >> 4, q4 = (idx & 15) << 2;
      tmp[i] = *(const float4*)(qbase + (size_t)r * CC + ch + q4);
    }
    #pragma unroll
    for (int i = 0; i < 8; ++i) {
      int idx = tid + i * 256;
      int r = idx >> 4, q4 = (idx & 15) << 2;
      *(float4*)&qt[r * LDSROW + q4] = tmp[i];
    }
#endif
    __syncthreads();
    const float* arow = &qt[(wave * 16 + m) * LDSROW];
    #pragma unroll
    for (int kk = 0; kk < KCH; kk += 4) {
      // one ds_load_b64 each (even k-offsets, 16B-aligned padded rows)
      v2f a = *(const v2f*)&arow[kk + kb];
      v2f bm = *(const v2f*)&bptr[ch + kk + kb];
      acc = __builtin_amdgcn_wmma_f32_16x16x4_f32(
          /*neg_a=*/false, a, /*neg_b=*/false, bm,
          /*c_mod=*/(short)0, acc, /*reuse_a=*/false, /*reuse_b=*/false);
    }
  }

  const float scale = 0.088388347648318447f;  // 1/sqrt(128)
  if (n < HH) {                               // lanes 0-3 (M 0-7), 16-19 (M 8-15)
    int mo = (lane >> 4) << 3;
    #pragma unroll
    for (int r = 0; r < 8; ++r) {
      int t = tbase + wave * 16 + mo + r;
      scores[((size_t)(b * HH + n)) * TT + t] = (acc[r] + c0_s[n]) * scale;
    }
  }
}

// ---------------------------------------------------------------------------
// Kernel 3: per (b,h): m = max_t scores, Z = sum_t exp(scores - m)
// ---------------------------------------------------------------------------
__global__ __launch_bounds__(256) void ts_stats_kernel(
    const float* __restrict__ scores, float* __restrict__ mz) {
  __shared__ float red[256];
  int bh = blockIdx.x, tid = threadIdx.x;
  const float* s = scores + (size_t)bh * TT;
  float mx = -INFINITY;
  for (int t = tid; t < TT; t += 256) mx = fmaxf(mx, s[t]);
  red[tid] = mx;
  __syncthreads();
  for (int st = 128; st > 0; st >>= 1) {
    if (tid < st) red[tid] = fmaxf(red[tid], red[tid + st]);
    __syncthreads();
  }
  mx = red[0];
  __syncthreads();
  float z = 0.f;
  for (int t = tid; t < TT; t += 256) z += __expf(s[t] - mx);
  red[tid] = z;
  __syncthreads();
  for (int st = 128; st > 0; st >>= 1) {
    if (tid < st) red[tid] += red[tid + st];
    __syncthreads();
  }
  if (tid == 0) { mz[bh * 2] = mx; mz[bh * 2 + 1] = red[0]; }
}

// ---------------------------------------------------------------------------
// Kernel 4: w[b,t] = mean_h softmax(scores)[b,h,t]
// ---------------------------------------------------------------------------
__global__ __launch_bounds__(256) void ts_weights_kernel(
    const float* __restrict__ scores, const float* __restrict__ mz,
    float* __restrict__ w) {
  int g = blockIdx.x * 256 + threadIdx.x;  // 0 .. B*T-1
  int b = g >> 12, t = g & (TT - 1);
  float acc = 0.f;
  #pragma unroll
  for (int h = 0; h < HH; ++h) {
    int bh = b * HH + h;
    acc += __expf(scores[(size_t)bh * TT + t] - mz[bh * 2]) / mz[bh * 2 + 1];
  }
  w[g] = acc * 0.25f;
}

// ---------------------------------------------------------------------------
// Kernel 5: per-batch top-k (iterative argmax, tie -> lower index, matching
// jax.lax.top_k's descending order) + dual gather of query/value rows.
// ---------------------------------------------------------------------------
__global__ __launch_bounds__(256) void ts_topk_gather_kernel(
    const float* __restrict__ w, const float* __restrict__ query,
    const float* __restrict__ value, const int* __restrict__ topk_p,
    float* __restrict__ out) {
  __shared__ float wv[TT];
  __shared__ float rv[256];
  __shared__ int ri[256];
  __shared__ int sel[64];
  int b = blockIdx.x, tid = threadIdx.x;
  int kt = *topk_p;
  if (kt > 64) kt = 64;
  for (int t = tid; t < TT; t += 256) wv[t] = w[(size_t)b * TT + t];
  __syncthreads();
  for (int it = 0; it < kt; ++it) {
    float bv = -INFINITY;
    int bi = 0;
    for (int t = tid; t < TT; t += 256) {
      float v = wv[t];
      if (v > bv) { bv = v; bi = t; }   // ascending scan keeps lowest index on tie
    }
    rv[tid] = bv; ri[tid] = bi;
    __syncthreads();
    for (int st = 128; st > 0; st >>= 1) {
      if (tid < st) {
        if (rv[tid + st] > rv[tid] ||
            (rv[tid + st] == rv[tid] && ri[tid + st] < ri[tid])) {
          rv[tid] = rv[tid + st];
          ri[tid] = ri[tid + st];
        }
      }
      __syncthreads();
    }
    if (tid == 0) { sel[it] = ri[0]; wv[ri[0]] = -INFINITY; }
    __syncthreads();
  }
  // gather: out = [top_k_query (B,kt,C) | top_k_value (B,kt,C)]
  size_t qout = (size_t)b * kt * CC;
  size_t vout = (size_t)BB * kt * CC + qout;
  for (int e = tid; e < kt * CC; e += 256) {
    int r = e >> 9, c = e & (CC - 1);
    int srow = sel[r];
    size_t src = ((size_t)b * TT + srow) * CC + c;
    out[qout + e] = query[src];
    out[vout + e] = value[src];
  }
}

// ---------------------------------------------------------------------------
extern "C" void kernel_launch(void* const* d_in, const int* in_sizes, int n_in,
                              void* d_out, int out_size, void* d_ws,
                              size_t ws_size, hipStream_t stream) {
  const float* query = (const float*)d_in[0];
  const float* key   = (const float*)d_in[1];
  const float* value = (const float*)d_in[2];
  const float* w_q   = (const float*)d_in[3];
  const float* b_q   = (const float*)d_in[4];
  const float* w_k   = (const float*)d_in[5];
  const float* b_k   = (const float*)d_in[6];
  const int*   topk  = (const int*)d_in[7];
  float* out = (float*)d_out;

  // workspace layout (floats): vq | c0 | scores | mz | w   (~2.8 MB total)
  float* vq = (float*)d_ws;
  float* c0 = vq + (size_t)BB * HH * CC;          // 65536
  float* sc = c0 + BB * HH;                       // +128
  float* mz = sc + (size_t)BB * HH * TT;          // +524288
  float* wt = mz + BB * HH * 2;                   // +256

  ts_proj_kernel<<<BB, 256, 0, stream>>>(key, w_q, b_q, w_k, b_k, vq, c0);
  ts_scores_kernel<<<BB * (TT / ROWS), 256, 0, stream>>>(query, vq, c0, sc);
  ts_stats_kernel<<<BB * HH, 256, 0, stream>>>(sc, mz);
  ts_weights_kernel<<<BB * TT / 256, 256, 0, stream>>>(sc, mz, wt);
  ts_topk_gather_kernel<<<BB, 256, 0, stream>>>(wt, query, value, topk, out);
}